// GCN_24644522345229
// MI455X (gfx1250) — compile-verified
//
#include <hip/hip_runtime.h>

typedef __attribute__((ext_vector_type(2))) float v2f;
typedef __attribute__((ext_vector_type(8))) float v8f;

#define IN_DIM 256
#define HID    128
#define OUTD   32
#define LDS_PAD 4   // break 16-way LDS bank conflicts on row-strided reads

#if __has_builtin(__builtin_amdgcn_global_load_async_to_lds_b128)
#define HAVE_ASYNC_LDS 1
typedef __attribute__((__vector_size__(4 * sizeof(int)))) int v4i;
typedef v4i* v4i_p;                                        // generic, per diagnostic
typedef __attribute__((address_space(3))) v4i* as3_v4i_p;  // LDS side
#else
#define HAVE_ASYNC_LDS 0
#endif

// -------------------------------------------------------------------------
// GEMM1: h0 = x @ W1 + b1      x:[M,256] f32, W1:[256,128], h0:[M,128]
// Block = 256 threads = 8 waves. Block stages one 16-row A tile in LDS via
// GLOBAL_LOAD_ASYNC_TO_LDS_B128 (ASYNCcnt path); wave w computes N-tile w
// with V_WMMA_F32_16X16X4_F32 (exact fp32, matches reference precision).
// -------------------------------------------------------------------------
__global__ __launch_bounds__(256) void gemm1_wmma(
    const float* __restrict__ x, const float* __restrict__ W1,
    const float* __restrict__ b1, float* __restrict__ h0, int M)
{
    __shared__ float lx[16 * (IN_DIM + LDS_PAD)];
    const int m0 = blockIdx.x * 16;
    const bool full = (m0 + 16 <= M);   // uniform per block; always true for M%16==0

    if (full) {
#if HAVE_ASYNC_LDS
        // 16x256 floats = 1024 x b128 chunks; 4 async issues per thread,
        // memory -> LDS directly, no VGPR round trip. Tracked by ASYNCcnt.
        #pragma unroll
        for (int t = 0; t < 4; ++t) {
            const int ch = threadIdx.x + t * 256;   // b128 chunk id, 0..1023
            const int r  = ch >> 6;                 // row 0..15
            const int c4 = (ch & 63) * 4;           // col 0..252 step 4
            __builtin_amdgcn_global_load_async_to_lds_b128(
                (v4i_p)(x + (size_t)(m0 + r) * IN_DIM + c4),
                (as3_v4i_p)(lx + r * (IN_DIM + LDS_PAD) + c4),
                0, 0);
        }
        asm volatile("s_wait_asynccnt 0" ::: "memory");
#else
        #pragma unroll
        for (int t = 0; t < 4; ++t) {
            const int ch = threadIdx.x + t * 256;
            const int r  = ch >> 6;
            const int c4 = (ch & 63) * 4;
            *(float4*)(lx + r * (IN_DIM + LDS_PAD) + c4) =
                *(const float4*)(x + (size_t)(m0 + r) * IN_DIM + c4);
        }
#endif
    } else {
        for (int i = threadIdx.x; i < 16 * IN_DIM; i += 256) {
            int r = i >> 8, c = i & (IN_DIM - 1);
            int rr = m0 + r; if (rr >= M) rr = M - 1;
            lx[r * (IN_DIM + LDS_PAD) + c] = x[(size_t)rr * IN_DIM + c];
        }
    }
    __syncthreads();

    const int lane = threadIdx.x & 31;
    const int half = lane >> 4;        // 0: lanes 0-15, 1: lanes 16-31
    const int lm   = lane & 15;
    const int n0   = (threadIdx.x >> 5) * 16;   // wave -> N tile

    v8f c = {};
    for (int k = 0; k < IN_DIM; k += 4) {
        const int ka = k + 2 * half;   // half 0 -> K={k,k+1}, half 1 -> K={k+2,k+3}
        v2f a;                          // A 16x4: lane lm = row, 2 K vals / lane
        a.x = lx[lm * (IN_DIM + LDS_PAD) + ka];
        a.y = lx[lm * (IN_DIM + LDS_PAD) + ka + 1];
        v2f b;                          // B 4x16: lane lm = col n0+lm
        b.x = W1[(size_t)ka       * HID + n0 + lm];
        b.y = W1[(size_t)(ka + 1) * HID + n0 + lm];
        c = __builtin_amdgcn_wmma_f32_16x16x4_f32(
                false, a, false, b, (short)0, c, false, false);
    }

    const float bias = b1[n0 + lm];
    const int rbase = m0 + 8 * half;   // C/D: VGPR r = row (8*half + r)
    if (full) {
        float* p = h0 + (size_t)rbase * HID + n0 + lm;
        #pragma unroll
        for (int r = 0; r < 8; ++r)
            p[(size_t)r * HID] = c[r] + bias;
    } else {
        #pragma unroll
        for (int r = 0; r < 8; ++r) {
            int row = rbase + r;
            if (row < M) h0[(size_t)row * HID + n0 + lm] = c[r] + bias;
        }
    }
}

// -------------------------------------------------------------------------
// GEMM2: h2 = relu(h1) @ W2 + b2   (ReLU fused into LDS staging, so staging
// stays on the VGPR path here). Block = 64 threads = 2 waves (N=32).
// -------------------------------------------------------------------------
__global__ __launch_bounds__(64) void gemm2_relu_wmma(
    const float* __restrict__ h1, const float* __restrict__ W2,
    const float* __restrict__ b2, float* __restrict__ h2, int M)
{
    __shared__ float lh[16 * (HID + LDS_PAD)];
    const int m0 = blockIdx.x * 16;
    const bool full = (m0 + 16 <= M);

    if (full) {
        // 16x128 floats = 512 float4 chunks; 8 per thread, ReLU on the fly.
        #pragma unroll
        for (int t = 0; t < 8; ++t) {
            const int ch = threadIdx.x + t * 64;    // 0..511
            const int r  = ch >> 5;                 // row 0..15
            const int c4 = (ch & 31) * 4;           // col 0..124 step 4
            float4 v = *(const float4*)(h1 + (size_t)(m0 + r) * HID + c4);
            v.x = fmaxf(v.x, 0.0f); v.y = fmaxf(v.y, 0.0f);
            v.z = fmaxf(v.z, 0.0f); v.w = fmaxf(v.w, 0.0f);
            *(float4*)(lh + r * (HID + LDS_PAD) + c4) = v;
        }
    } else {
        for (int i = threadIdx.x; i < 16 * HID; i += 64) {
            int r = i >> 7, c = i & (HID - 1);
            int rr = m0 + r; if (rr >= M) rr = M - 1;
            lh[r * (HID + LDS_PAD) + c] = fmaxf(h1[(size_t)rr * HID + c], 0.0f);
        }
    }
    __syncthreads();

    const int lane = threadIdx.x & 31;
    const int half = lane >> 4;
    const int lm   = lane & 15;
    const int n0   = (threadIdx.x >> 5) * 16;

    v8f c = {};
    for (int k = 0; k < HID; k += 4) {
        const int ka = k + 2 * half;
        v2f a;
        a.x = lh[lm * (HID + LDS_PAD) + ka];
        a.y = lh[lm * (HID + LDS_PAD) + ka + 1];
        v2f b;
        b.x = W2[(size_t)ka       * OUTD + n0 + lm];
        b.y = W2[(size_t)(ka + 1) * OUTD + n0 + lm];
        c = __builtin_amdgcn_wmma_f32_16x16x4_f32(
                false, a, false, b, (short)0, c, false, false);
    }

    const float bias = b2[n0 + lm];
    const int rbase = m0 + 8 * half;
    if (full) {
        float* p = h2 + (size_t)rbase * OUTD + n0 + lm;
        #pragma unroll
        for (int r = 0; r < 8; ++r)
            p[(size_t)r * OUTD] = c[r] + bias;
    } else {
        #pragma unroll
        for (int r = 0; r < 8; ++r) {
            int row = rbase + r;
            if (row < M) h2[(size_t)row * OUTD + n0 + lm] = c[r] + bias;
        }
    }
}

// -------------------------------------------------------------------------
// SpMM layer 1: h1[row] += val * h0[col], 128 features.
// One wave per edge (edge index uniform per wave -> scalarized loads);
// each lane handles 4 consecutive floats. h0/h1 (51 MB each) are L2-resident
// on MI455X (192 MB L2), so the float atomics resolve at L2 rate.
// -------------------------------------------------------------------------
__global__ __launch_bounds__(256) void spmm_h128(
    const int* __restrict__ erow, const int* __restrict__ ecol,
    const float* __restrict__ eval, const float* __restrict__ h0,
    float* __restrict__ h1, int E)
{
    const long long t = (long long)blockIdx.x * 256 + threadIdx.x;
    const long long e = t >> 5;
    if (e >= E) return;
    const int lane = (int)(t & 31);

    const int   r    = erow[e];
    const int   cidx = ecol[e];
    const float v    = eval[e];

    const float4 m = *(const float4*)(h0 + (size_t)cidx * HID + lane * 4);
    float* dst = h1 + (size_t)r * HID + lane * 4;
    atomicAdd(dst + 0, v * m.x);
    atomicAdd(dst + 1, v * m.y);
    atomicAdd(dst + 2, v * m.z);
    atomicAdd(dst + 3, v * m.w);
}

// -------------------------------------------------------------------------
// SpMM layer 2: out[row] += val * h2[col], 32 features. One wave per edge,
// one float per lane.
// -------------------------------------------------------------------------
__global__ __launch_bounds__(256) void spmm_h32(
    const int* __restrict__ erow, const int* __restrict__ ecol,
    const float* __restrict__ eval, const float* __restrict__ h2,
    float* __restrict__ out, int E)
{
    const long long t = (long long)blockIdx.x * 256 + threadIdx.x;
    const long long e = t >> 5;
    if (e >= E) return;
    const int lane = (int)(t & 31);

    const int   r    = erow[e];
    const int   cidx = ecol[e];
    const float v    = eval[e];

    atomicAdd(out + (size_t)r * OUTD + lane, v * h2[(size_t)cidx * OUTD + lane]);
}

__global__ __launch_bounds__(256) void zero_f32(float4* __restrict__ p, size_t n4)
{
    size_t i = (size_t)blockIdx.x * 256 + threadIdx.x;
    size_t stride = (size_t)gridDim.x * 256;
    const float4 z = {0.0f, 0.0f, 0.0f, 0.0f};
    for (; i < n4; i += stride) p[i] = z;
}

extern "C" void kernel_launch(void* const* d_in, const int* in_sizes, int n_in,
                              void* d_out, int out_size, void* d_ws, size_t ws_size,
                              hipStream_t stream)
{
    const float* x    = (const float*)d_in[0];
    const int*   erow = (const int*)  d_in[1];
    const int*   ecol = (const int*)  d_in[2];
    const float* eval = (const float*)d_in[3];
    const float* W1   = (const float*)d_in[4];
    const float* b1   = (const float*)d_in[5];
    const float* W2   = (const float*)d_in[6];
    const float* b2   = (const float*)d_in[7];

    const int M = in_sizes[0] / IN_DIM;   // 100000 nodes
    const int E = in_sizes[1];            // 1600000 edges

    float* h0 = (float*)d_ws;                       // [M,128]  51.2 MB
    float* h1 = h0 + (size_t)M * HID;               // [M,128]  51.2 MB
    float* h2 = h1 + (size_t)M * HID;               // [M,32]   12.8 MB
    float* out = (float*)d_out;

    const int mtiles = (M + 15) / 16;               // 6250

    // Layer 1 dense: h0 = x @ W1 + b1 (f32 WMMA, async-LDS staged A)
    gemm1_wmma<<<mtiles, 256, 0, stream>>>(x, W1, b1, h0, M);

    // Layer 1 sparse: h1 = SpMM(A, h0)
    zero_f32<<<2048, 256, 0, stream>>>((float4*)h1, (size_t)M * HID / 4);
    {
        long long threads = (long long)E * 32;
        int blocks = (int)((threads + 255) / 256);
        spmm_h128<<<blocks, 256, 0, stream>>>(erow, ecol, eval, h0, h1, E);
    }

    // Layer 2 dense: h2 = relu(h1) @ W2 + b2 (f32 WMMA, ReLU fused)
    gemm2_relu_wmma<<<mtiles, 64, 0, stream>>>(h1, W2, b2, h2, M);

    // Layer 2 sparse: out = SpMM(A, h2)
    zero_f32<<<2048, 256, 0, stream>>>((float4*)out, (size_t)M * OUTD / 4);
    {
        long long threads = (long long)E * 32;
        int blocks = (int)((threads + 255) / 256);
        spmm_h32<<<blocks, 256, 0, stream>>>(erow, ecol, eval, h2, out, E);
    }
}